// MoELayer_12936441495906
// MI455X (gfx1250) — compile-verified
//
#include <hip/hip_runtime.h>
#include <math.h>

// ---------------- problem constants ----------------
#define NTOK   4096          // B*T
#define HDIM   1024
#define FDIM   4096
#define NEXP   8
#define CAP    640           // int(N/E * 1.25)
#define LSTRIDE 4096         // per-group token-list stride (group 8 = fallback)
#define BSTR   272           // B LDS stride in float2 units (272 % 32 == 16 -> bank split)

typedef __attribute__((ext_vector_type(2))) float v2f;
typedef __attribute__((ext_vector_type(8))) float v8f;
typedef int v2i32 __attribute__((ext_vector_type(2)));

#if __has_builtin(__builtin_amdgcn_global_load_async_to_lds_b32) && \
    __has_builtin(__builtin_amdgcn_global_load_async_to_lds_b64)
#define USE_ASYNC 1
#else
#define USE_ASYNC 0
#endif

#if USE_ASYNC
typedef __attribute__((address_space(1))) int   gint;
typedef __attribute__((address_space(3))) int   lint;
typedef __attribute__((address_space(1))) v2i32 gint2;
typedef __attribute__((address_space(3))) v2i32 lint2;

__device__ __forceinline__ void async_b32(const float* g, float* l) {
    float* gn = const_cast<float*>(g);
    __builtin_amdgcn_global_load_async_to_lds_b32((gint*)gn, (lint*)l, 0, 0);
}
__device__ __forceinline__ void async_b64(const float* g, float* l) {
    float* gn = const_cast<float*>(g);
    __builtin_amdgcn_global_load_async_to_lds_b64((gint2*)gn, (lint2*)l, 0, 0);
}
#endif

__device__ __forceinline__ void wait_async0() {
#if __has_builtin(__builtin_amdgcn_s_wait_asynccnt)
    __builtin_amdgcn_s_wait_asynccnt(0);
#else
    asm volatile("s_wait_asynccnt 0x0" ::: "memory");
#endif
}

// ---------------- d_out layout (floats) ----------------
#define OFF_Y      0
#define OFF_UTIL   4194304
#define OFF_F      4194312
#define OFF_P      4194320
#define OFF_LOGITS 4194328
#define OFF_GTOP   4227096
#define OFF_IDX    4259864

// =====================================================================
// Kernel 1: router — one wave (32 lanes) per token
// =====================================================================
__global__ void router_kernel(const float* __restrict__ x,
                              const float* __restrict__ Wr,
                              const float* __restrict__ br,
                              float* __restrict__ logits_out,
                              float* __restrict__ gtop_out,
                              float* __restrict__ idxf_out,
                              int*   __restrict__ idx_ws) {
    int wid  = (blockIdx.x * blockDim.x + threadIdx.x) >> 5;   // token
    int lane = threadIdx.x & 31;
    if (wid >= NTOK) return;

    const float* xr = x + (size_t)wid * HDIM;
    float acc[NEXP];
#pragma unroll
    for (int e = 0; e < NEXP; ++e) acc[e] = 0.f;

    for (int j = 0; j < HDIM / 32; ++j) {
        int r = j * 32 + lane;
        float xv = xr[r];
        const float* wr = Wr + (size_t)r * NEXP;
        float4 w0 = *(const float4*)(wr);
        float4 w1 = *(const float4*)(wr + 4);
        acc[0] += xv * w0.x; acc[1] += xv * w0.y;
        acc[2] += xv * w0.z; acc[3] += xv * w0.w;
        acc[4] += xv * w1.x; acc[5] += xv * w1.y;
        acc[6] += xv * w1.z; acc[7] += xv * w1.w;
    }
#pragma unroll
    for (int e = 0; e < NEXP; ++e)
        for (int off = 16; off; off >>= 1)
            acc[e] += __shfl_xor(acc[e], off, 32);

    float lg[NEXP], gs[NEXP];
    float m = -1e30f;
#pragma unroll
    for (int e = 0; e < NEXP; ++e) { lg[e] = acc[e] + br[e]; m = fmaxf(m, lg[e]); }
    float s = 0.f;
#pragma unroll
    for (int e = 0; e < NEXP; ++e) { gs[e] = __expf(lg[e] - m); s += gs[e]; }
    float inv = 1.f / s;
#pragma unroll
    for (int e = 0; e < NEXP; ++e) gs[e] *= inv;

    // top-2, lower index wins ties (strict >)
    int i0 = 0;
#pragma unroll
    for (int e = 1; e < NEXP; ++e) if (gs[e] > gs[i0]) i0 = e;
    int i1 = (i0 == 0) ? 1 : 0;
#pragma unroll
    for (int e = 0; e < NEXP; ++e) if (e != i0 && gs[e] > gs[i1]) i1 = e;

    float denom = gs[i0] + gs[i1] + 1e-9f;
    if (lane < NEXP) {
        logits_out[wid * NEXP + lane] = lg[lane];
        float gt = (lane == i0 || lane == i1) ? gs[lane] / denom : 0.f;
        gtop_out[wid * NEXP + lane] = gt;
    }
    if (lane == 0) {
        idx_ws[wid * 2 + 0] = i0;
        idx_ws[wid * 2 + 1] = i1;
        idxf_out[wid * 2 + 0] = (float)i0;
        idxf_out[wid * 2 + 1] = (float)i1;
    }
}

// =====================================================================
// Kernel 2: capacity assignment — single block, 1024 threads, 4 tok/thread
// =====================================================================
typedef unsigned long long ull;

__device__ __forceinline__ void packadd(ull& lo, ull& hi, int e) {
    if (e < 4) lo += 1ull << (16 * e);
    else       hi += 1ull << (16 * (e - 4));
}
__device__ __forceinline__ int packget(ull lo, ull hi, int e) {
    return (int)(((e < 4) ? (lo >> (16 * e)) : (hi >> (16 * (e - 4)))) & 0xFFFF);
}

__global__ __launch_bounds__(1024)
void assign_kernel(const int* __restrict__ idx_ws,
                   const float* __restrict__ gtop_out,
                   int* __restrict__ lists,
                   int* __restrict__ counts,
                   float* __restrict__ util_out,
                   float* __restrict__ f_out,
                   float* __restrict__ P_out) {
    __shared__ ull sbuf[4096];            // 32 KB, reused as float[8192] at the end
    __shared__ int fb_cnt;
    ull* aLo = sbuf;        ull* aHi = sbuf + 1024;
    ull* bLo = sbuf + 2048; ull* bHi = sbuf + 3072;

    int tid = threadIdx.x;
    if (tid == 0) fb_cnt = 0;
    int base = tid * 4;

    int p[4], s[4];
#pragma unroll
    for (int j = 0; j < 4; ++j) {
        p[j] = idx_ws[(base + j) * 2 + 0];
        s[j] = idx_ws[(base + j) * 2 + 1];
    }

    // ---- pass 1: primary ranks ----
    ull lo = 0, hi = 0;
#pragma unroll
    for (int j = 0; j < 4; ++j) packadd(lo, hi, p[j]);

    __syncthreads();
    aLo[tid] = lo; aHi[tid] = hi;
    __syncthreads();
    ull *cl = aLo, *ch = aHi, *nl = bLo, *nh = bHi;
    for (int off = 1; off < 1024; off <<= 1) {
        ull l = cl[tid], h = ch[tid];
        if (tid >= off) { l += cl[tid - off]; h += ch[tid - off]; }
        nl[tid] = l; nh[tid] = h;
        __syncthreads();
        ull* t;
        t = cl; cl = nl; nl = t;
        t = ch; ch = nh; nh = t;
    }
    ull exLo = cl[tid] - lo, exHi = ch[tid] - hi;
    ull tpLo = cl[1023],     tpHi = ch[1023];

    int c8[NEXP];
#pragma unroll
    for (int e = 0; e < NEXP; ++e) c8[e] = packget(exLo, exHi, e);
    bool keep1[4];
#pragma unroll
    for (int j = 0; j < 4; ++j) {
        int e = p[j];
        int rank = ++c8[e];
        keep1[j] = (rank <= CAP);
        if (keep1[j]) lists[e * LSTRIDE + rank - 1] = base + j;
    }
    int tp[NEXP], used[NEXP];
#pragma unroll
    for (int e = 0; e < NEXP; ++e) {
        tp[e] = packget(tpLo, tpHi, e);
        used[e] = tp[e] < CAP ? tp[e] : CAP;
    }

    // ---- pass 2: secondary ranks among overflow tokens ----
    lo = 0; hi = 0;
#pragma unroll
    for (int j = 0; j < 4; ++j) if (!keep1[j]) packadd(lo, hi, s[j]);

    __syncthreads();
    aLo[tid] = lo; aHi[tid] = hi;
    __syncthreads();
    cl = aLo; ch = aHi; nl = bLo; nh = bHi;
    for (int off = 1; off < 1024; off <<= 1) {
        ull l = cl[tid], h = ch[tid];
        if (tid >= off) { l += cl[tid - off]; h += ch[tid - off]; }
        nl[tid] = l; nh[tid] = h;
        __syncthreads();
        ull* t;
        t = cl; cl = nl; nl = t;
        t = ch; ch = nh; nh = t;
    }
    exLo = cl[tid] - lo; exHi = ch[tid] - hi;
    ull tsLo = cl[1023]; ull tsHi = ch[1023];

#pragma unroll
    for (int e = 0; e < NEXP; ++e) c8[e] = packget(exLo, exHi, e);
#pragma unroll
    for (int j = 0; j < 4; ++j) {
        if (keep1[j]) continue;
        int e = s[j];
        int rank = (++c8[e]) + used[e];
        if (rank <= CAP) {
            lists[e * LSTRIDE + rank - 1] = base + j;
        } else {
            int pos = atomicAdd(&fb_cnt, 1);
            lists[NEXP * LSTRIDE + pos] = base + j;   // fallback list
        }
    }
    __syncthreads();

    if (tid == 0) {
        counts[NEXP] = fb_cnt;
#pragma unroll
        for (int e = 0; e < NEXP; ++e) {
            int ts = packget(tsLo, tsHi, e);
            int room = CAP - used[e];
            int fin = used[e] + (ts < room ? ts : room);
            counts[e] = fin;
            util_out[e] = (float)fin / (float)NTOK;
            f_out[e]    = (float)tp[e] / (float)NTOK;
        }
    }

    // ---- P = g_top.mean(0) : tree reduction reusing sbuf as float ----
    float pacc[NEXP];
#pragma unroll
    for (int e = 0; e < NEXP; ++e) pacc[e] = 0.f;
#pragma unroll
    for (int j = 0; j < 4; ++j)
#pragma unroll
        for (int e = 0; e < NEXP; ++e)
            pacc[e] += gtop_out[(base + j) * NEXP + e];
    __syncthreads();
    float* fs = (float*)sbuf;                     // 8192 floats
#pragma unroll
    for (int e = 0; e < NEXP; ++e) fs[e * 1024 + tid] = pacc[e];
    __syncthreads();
    for (int off = 512; off; off >>= 1) {
        if (tid < off)
#pragma unroll
            for (int e = 0; e < NEXP; ++e)
                fs[e * 1024 + tid] += fs[e * 1024 + tid + off];
        __syncthreads();
    }
    if (tid < NEXP) P_out[tid] = fs[tid * 1024] / (float)NTOK;
}

// =====================================================================
// Kernels 3/4: gathered expert GEMM, fp32 WMMA 16x16x4
//   C-tile per WG: 32 (M) x 256 (cols); 8 waves, each 16x64 (4 v8f accs)
//   Double-buffered LDS; B staged K-pair-interleaved (fragment = 1 b64 load)
//   Staging uses CDNA5 async global->LDS when available (ASYNCcnt).
// =====================================================================
template <int DO_GELU>
__global__ __launch_bounds__(256)
void moe_gemm(const float* __restrict__ A,      // row stride K (x or h_buf)
              const float* __restrict__ Wexp,   // (E, K, NC)
              const float* __restrict__ bexp,   // (E, NC)
              const float* __restrict__ Wfb,    // (K, NC)
              const float* __restrict__ bfb,    // (NC)
              const int*   __restrict__ lists,
              const int*   __restrict__ counts,
              float*       __restrict__ Out,    // row stride NC, indexed by token
              int K, int NC) {
    __shared__ float As[2][32][20];      // padded stride 20 -> conflict-free A frags
    __shared__ v2f   Bs[2][8 * BSTR];    // [buf][kpair*BSTR + col], pair-interleaved

    int g   = blockIdx.z;
    int cnt = counts[g];
    int m0  = blockIdx.y * 32;
    if (m0 >= cnt) return;

    const float* B    = (g < NEXP) ? (Wexp + (size_t)g * K * NC) : Wfb;
    const float* bias = (g < NEXP) ? (bexp + (size_t)g * NC)     : bfb;
    const int*   list = lists + g * LSTRIDE;
    int fblk = blockIdx.x * 256;

    int tid  = threadIdx.x;
    int lane = tid & 31;
    int wave = tid >> 5;
    int mh   = wave >> 2;        // 0/1 : which 16-row half
    int fsub = wave & 3;         // 0..3: which 64-col slab
    int l16  = lane & 15;
    int hl   = lane >> 4;        // 0/1

    // A staging: 32 rows x 16 cols; thread -> (row, 2 cols)
    int ar = tid >> 3;           // 0..31
    int ac = (tid & 7) * 2;      // 0..14
    int arow = m0 + ar;
    int atok = (arow < cnt) ? list[arow] : list[m0];   // clamp to a valid token
    const float* Aptr = A + (size_t)atok * K + ac;

    // B staging: each thread owns one column, walks 16 K-rows per step
    int bcol = wave * 32 + lane;             // 0..255
    const float* Bcp = B + fblk + bcol;

    v8f cacc[4];
#pragma unroll
    for (int n = 0; n < 4; ++n) cacc[n] = (v8f){0.f,0.f,0.f,0.f,0.f,0.f,0.f,0.f};

    int nsteps = K / 16;

#if USE_ASYNC
    // ---- async prologue: stage step 0 into buf 0 ----
    async_b64(Aptr, &As[0][ar][ac]);
#pragma unroll
    for (int k = 0; k < 16; ++k) {
        float* dst = (float*)&Bs[0][(k >> 1) * BSTR + bcol] + (k & 1);
        async_b32(Bcp + (size_t)k * NC, dst);
    }
    wait_async0();
    __syncthreads();
#else
    {
        v2f av = *(const v2f*)(Aptr);
        float bv[16];
#pragma unroll
        for (int k = 0; k < 16; ++k) bv[k] = Bcp[(size_t)k * NC];
        As[0][ar][ac] = av.x; As[0][ar][ac + 1] = av.y;
#pragma unroll
        for (int k2 = 0; k2 < 8; ++k2)
            Bs[0][k2 * BSTR + bcol] = (v2f){bv[2 * k2], bv[2 * k2 + 1]};
    }
    __syncthreads();
#endif

    int p = 0;
    for (int st = 0; st < nsteps; ++st) {
        int k0n = (st + 1) * 16;
        bool more = (st + 1 < nsteps);

#if USE_ASYNC
        if (more) {
            async_b64(Aptr + k0n, &As[p ^ 1][ar][ac]);
#pragma unroll
            for (int k = 0; k < 16; ++k) {
                float* dst = (float*)&Bs[p ^ 1][(k >> 1) * BSTR + bcol] + (k & 1);
                async_b32(Bcp + (size_t)(k0n + k) * NC, dst);
            }
        }
#else
        v2f an;
        float bn[16];
        if (more) {
            an = *(const v2f*)(Aptr + k0n);
#pragma unroll
            for (int k = 0; k < 16; ++k) bn[k] = Bcp[(size_t)(k0n + k) * NC];
        }
#endif

        // ---- compute from buffer p ----
#pragma unroll
        for (int k4 = 0; k4 < 4; ++k4) {
            // A 16x4 f32 frag: VGPR0 = K0(lanes0-15)/K2(lanes16-31), VGPR1 = K1/K3
            v2f a = *(const v2f*)&As[p][mh * 16 + l16][k4 * 4 + hl * 2];
#pragma unroll
            for (int n = 0; n < 4; ++n) {
                int col = fsub * 64 + n * 16 + l16;
                v2f b = Bs[p][(k4 * 2 + hl) * BSTR + col];   // one b64 load
                cacc[n] = __builtin_amdgcn_wmma_f32_16x16x4_f32(
                    false, a, false, b, (short)0, cacc[n], false, false);
            }
        }

#if USE_ASYNC
        if (more) wait_async0();
        __syncthreads();
#else
        if (more) {
            As[p ^ 1][ar][ac] = an.x; As[p ^ 1][ar][ac + 1] = an.y;
#pragma unroll
            for (int k2 = 0; k2 < 8; ++k2)
                Bs[p ^ 1][k2 * BSTR + bcol] = (v2f){bn[2 * k2], bn[2 * k2 + 1]};
        }
        __syncthreads();
#endif
        p ^= 1;
    }

    // epilogue: bias (+ GeLU) and scatter by token id
#pragma unroll
    for (int n = 0; n < 4; ++n) {
#pragma unroll
        for (int v = 0; v < 8; ++v) {
            int M   = v + 8 * hl;                 // row within 16-tile (C layout)
            int row = m0 + mh * 16 + M;
            if (row < cnt) {
                int tok = list[row];
                int col = fblk + fsub * 64 + n * 16 + l16;
                float val = cacc[n][v] + bias[col];
                if (DO_GELU)
                    val = 0.5f * val * (1.0f + erff(val * 0.70710678118654752f));
                Out[(size_t)tok * NC + col] = val;
            }
        }
    }
}

// =====================================================================
// launch
// =====================================================================
extern "C" void kernel_launch(void* const* d_in, const int* in_sizes, int n_in,
                              void* d_out, int out_size, void* d_ws, size_t ws_size,
                              hipStream_t stream) {
    const float* x   = (const float*)d_in[0];
    const float* Wr  = (const float*)d_in[1];
    const float* br  = (const float*)d_in[2];
    const float* W1  = (const float*)d_in[3];
    const float* b1  = (const float*)d_in[4];
    const float* W2  = (const float*)d_in[5];
    const float* b2  = (const float*)d_in[6];
    const float* Wf1 = (const float*)d_in[7];
    const float* bf1 = (const float*)d_in[8];
    const float* Wf2 = (const float*)d_in[9];
    const float* bf2 = (const float*)d_in[10];

    float* out = (float*)d_out;
    float* y_out      = out + OFF_Y;
    float* util_out   = out + OFF_UTIL;
    float* f_out      = out + OFF_F;
    float* P_out      = out + OFF_P;
    float* logits_out = out + OFF_LOGITS;
    float* gtop_out   = out + OFF_GTOP;
    float* idxf_out   = out + OFF_IDX;

    char* w = (char*)d_ws;
    int*   idx_ws = (int*)(w);                                   // N*2 ints
    int*   lists  = (int*)(w + 32768);                           // 9 * LSTRIDE ints
    int*   counts = (int*)(w + 32768 + 9 * LSTRIDE * 4);         // 16 ints
    float* h_buf  = (float*)(w + 262144);                        // N*F floats

    // 1) router: 4096 waves = 512 blocks x 256 threads
    router_kernel<<<dim3((NTOK * 32) / 256), dim3(256), 0, stream>>>(
        x, Wr, br, logits_out, gtop_out, idxf_out, idx_ws);

    // 2) capacity assignment: single 1024-thread block
    assign_kernel<<<dim3(1), dim3(1024), 0, stream>>>(
        idx_ws, gtop_out, lists, counts, util_out, f_out, P_out);

    // 3) up-projection + GeLU -> h_buf[token][F]
    moe_gemm<1><<<dim3(FDIM / 256, NTOK / 32, NEXP + 1), dim3(256), 0, stream>>>(
        x, W1, b1, Wf1, bf1, lists, counts, h_buf, HDIM, FDIM);

    // 4) down-projection -> y[token][H]
    moe_gemm<0><<<dim3(HDIM / 256, NTOK / 32, NEXP + 1), dim3(256), 0, stream>>>(
        h_buf, W2, b2, Wf2, bf2, lists, counts, y_out, FDIM, HDIM);
}